// ConvDS_8418135900870
// MI455X (gfx1250) — compile-verified
//
#include <hip/hip_runtime.h>

typedef __attribute__((ext_vector_type(2))) float v2f;
typedef __attribute__((ext_vector_type(8))) float v8f;

// Haar 2x2 pixel-unshuffle downsample fused into D = A(16x4) x B(4x16) WMMA.
//   A rows 0-3 and 8-11 hold the 4x4 Haar matrix W (rows 4-7 / 12-15 zero).
//   B column n holds pixel n's 2x2 block (a,b,c,d): half-wave 0 supplies K=0,1
//   from input row 2h, half-wave 1 supplies K=2,3 from input row 2h+1.
//   D[j,n] = out_j(pixel n), duplicated in rows j and j+8 so every lane holds
//   the full result; half 0 stores j=0,1 and half 1 stores j=2,3 (full-wave
//   coalesced stores). Memory-bound op: 256B in + 256B out per WMMA.
__global__ __launch_bounds__(256) void haar_ds_wmma(
    const float* __restrict__ x,      // [16,3,1024,1024]
    const float* __restrict__ cw,     // [4,4] Haar weights
    float* __restrict__ out) {        // [16,12,512,512]
  constexpr int W_IN = 1024;
  constexpr int W_OUT = 512;

  const int lane = threadIdx.x & 31;
  const int wave = threadIdx.x >> 5;   // 0..7
  const int n    = lane & 15;          // pixel slot == A-matrix row M
  const int half = lane >> 4;          // 0 -> K=0,1 (row 2h); 1 -> K=2,3 (row 2h+1)

  const int h     = blockIdx.x;        // output row, 0..511
  const int plane = blockIdx.y;        // b*3 + c, 0..47
  const int b = plane / 3;
  const int c = plane - 3 * b;

  // ---- Build the A operand (ISA 32-bit 16x4 layout: lane m holds row M=m,
  // .x = K=k0, .y = K=k0+1 with k0 = 2*half). Valid rows: 0-3 and 8-11.
  const int  k0    = half * 2;
  const bool valid = ((n & 4) == 0);   // n in {0..3, 8..11}
  const int  jrow  = n & 3;
  v2f amat;
  amat.x = valid ? cw[jrow * 4 + k0]     : 0.0f;
  amat.y = valid ? cw[jrow * 4 + k0 + 1] : 0.0f;

  // ---- Input row pointer for this lane's half (row 2h or 2h+1 of the plane).
  const float* rowp = x + (size_t)plane * ((size_t)W_IN * W_IN)
                        + (size_t)(2 * h + half) * W_IN;

  // ---- Output bases per Haar component j: channel = b*12 + j*3 + c.
  const size_t ob0 = ((size_t)(b * 12 + 0 * 3 + c) * W_OUT + h) * W_OUT;
  const size_t ob1 = ((size_t)(b * 12 + 1 * 3 + c) * W_OUT + h) * W_OUT;
  const size_t ob2 = ((size_t)(b * 12 + 2 * 3 + c) * W_OUT + h) * W_OUT;
  const size_t ob3 = ((size_t)(b * 12 + 3 * 3 + c) * W_OUT + h) * W_OUT;
  const size_t base_lo = half ? ob2 : ob0;   // half 0 stores j=0,1; half 1 -> j=2,3
  const size_t base_hi = half ? ob3 : ob1;

  const int px0 = wave * 64 + n;       // this wave owns output pixels [wave*64, wave*64+64)

#pragma unroll
  for (int it = 0; it < 4; ++it) {
    const int px = px0 + it * 16;
    // 2x2 block columns 2*px, 2*px+1 of this lane's input row: one b64 load.
    // Half-wave footprint = 128B contiguous -> fully coalesced.
    const v2f bmat = *(const v2f*)(rowp + 2 * px);

    v8f acc = {};
    // D = A x B + 0   (f32 in, f32 accumulate; matches reference precision)
    acc = __builtin_amdgcn_wmma_f32_16x16x4_f32(
        /*neg_a=*/false, amat, /*neg_b=*/false, bmat,
        /*c_mod=*/(short)0, acc, /*reuse_a=*/false, /*reuse_b=*/false);

    const float o_lo = half ? acc[2] : acc[0];
    const float o_hi = half ? acc[3] : acc[1];
    out[base_lo + px] = o_lo;
    out[base_hi + px] = o_hi;
  }
}

extern "C" void kernel_launch(void* const* d_in, const int* in_sizes, int n_in,
                              void* d_out, int out_size, void* d_ws, size_t ws_size,
                              hipStream_t stream) {
  (void)in_sizes; (void)n_in; (void)out_size; (void)d_ws; (void)ws_size;
  const float* x  = (const float*)d_in[0];   // [16,3,1024,1024] f32
  const float* cw = (const float*)d_in[1];   // [4,4] f32 Haar weights
  float* out = (float*)d_out;                // [16,12,512,512] f32

  // One block (8 waves x 64 px) per output row per (b,c) plane.
  dim3 grid(512, 48, 1);
  dim3 block(256, 1, 1);
  haar_ds_wmma<<<grid, block, 0, stream>>>(x, cw, out);
}